// FastfoodWrapper_80582176407620
// MI455X (gfx1250) — compile-verified
//
#include <hip/hip_runtime.h>
#include <hip/hip_bf16.h>

typedef __attribute__((ext_vector_type(2))) float v2f;
typedef __attribute__((ext_vector_type(8))) float v8f;

#define IN_F   4096
#define OUT_F  4096
#define BATCH  256
#define LLW    16777216   // 2^24
#define KT     32
#define XSTR   36         // LDS row stride (floats), mult of 4, conflict-friendly

#define IDX(i) ((i) + ((i) >> 6))   // bank-conflict padding for 4096-pt FWHT

// ---------------------------------------------------------------------------
// Kernel 1: seeds. Block 0: u_W = FWHT_1024(BB_W * v)  (first 16M FWHT collapses
// to this because the padded input is zero beyond 1024).
// Block 1: full bias path -> b_tot = b_init + FWHT_4096(u_b[Pi_b&1023]*GG_b)/div_b
// ---------------------------------------------------------------------------
__global__ __launch_bounds__(256)
void seed_kernel(const float* __restrict__ ld, const float* __restrict__ BBW,
                 float* __restrict__ uW,
                 const float* __restrict__ b_init, const float* __restrict__ BBb,
                 const float* __restrict__ GGb, const float* __restrict__ divb,
                 const int* __restrict__ Pib, float* __restrict__ btot) {
  __shared__ float u[1024];
  __shared__ float tb[4096];
  const int tid = threadIdx.x;

  if (blockIdx.x == 0) {
    for (int j = 0; j < 4; ++j) {
      int i = tid + 256 * j;
      u[i] = BBW[i] * ld[i];
    }
    __syncthreads();
    for (int h = 1; h < 1024; h <<= 1) {
      for (int p = tid; p < 512; p += 256) {
        int i0 = ((p & ~(h - 1)) << 1) | (p & (h - 1));
        int i1 = i0 + h;
        float a = u[i0], b = u[i1];
        u[i0] = a + b; u[i1] = a - b;
      }
      __syncthreads();
    }
    for (int j = 0; j < 4; ++j) {
      int i = tid + 256 * j;
      uW[i] = u[i];
    }
  } else {
    for (int j = 0; j < 4; ++j) {
      int i = tid + 256 * j;
      u[i] = BBb[i] * ld[i];    // only first 1024 of padded vec are nonzero
    }
    __syncthreads();
    for (int h = 1; h < 1024; h <<= 1) {
      for (int p = tid; p < 512; p += 256) {
        int i0 = ((p & ~(h - 1)) << 1) | (p & (h - 1));
        int i1 = i0 + h;
        float a = u[i0], b = u[i1];
        u[i0] = a + b; u[i1] = a - b;
      }
      __syncthreads();
    }
    // m_b[i] = u[i & 1023]; permute+scale
    for (int j = 0; j < 16; ++j) {
      int i = tid + 256 * j;
      tb[i] = u[Pib[i] & 1023] * GGb[i];
    }
    __syncthreads();
    for (int h = 1; h < 4096; h <<= 1) {
      for (int p = tid; p < 2048; p += 256) {
        int i0 = ((p & ~(h - 1)) << 1) | (p & (h - 1));
        int i1 = i0 + h;
        float a = tb[i0], b = tb[i1];
        tb[i0] = a + b; tb[i1] = a - b;
      }
      __syncthreads();
    }
    float inv = 1.0f / divb[0];
    for (int j = 0; j < 16; ++j) {
      int i = tid + 256 * j;
      btot[i] = b_init[i] + tb[i] * inv;
    }
  }
}

// ---------------------------------------------------------------------------
// Kernel 2: t[i] = u_W[Pi_W[i] & 1023] * GG_W[i]   (16M elements, vectorized)
// ---------------------------------------------------------------------------
__global__ __launch_bounds__(256)
void gather_kernel(const int* __restrict__ Pi, const float* __restrict__ GG,
                   const float* __restrict__ uW, float* __restrict__ t) {
  __shared__ float su[1024];
  const int tid = threadIdx.x;
  for (int j = 0; j < 4; ++j) su[tid + 256 * j] = uW[tid + 256 * j];
  __syncthreads();
  size_t base = ((size_t)blockIdx.x * 256 + tid) * 4;
  int4   pi = *(const int4*)&Pi[base];
  float4 g  = *(const float4*)&GG[base];
  float4 o;
  o.x = su[pi.x & 1023] * g.x;
  o.y = su[pi.y & 1023] * g.y;
  o.z = su[pi.z & 1023] * g.z;
  o.w = su[pi.w & 1023] * g.w;
  *(float4*)&t[base] = o;
}

// ---------------------------------------------------------------------------
// Kernel 3: FWHT stages over the low 12 bits: 4096 contiguous floats per block.
// ---------------------------------------------------------------------------
__global__ __launch_bounds__(256)
void fwht_contig_kernel(float* __restrict__ data) {
  __shared__ float s[4096 + 64];
  const int tid = threadIdx.x;
  float* base = data + (size_t)blockIdx.x * 4096;
  for (int j = 0; j < 4; ++j) {
    int e = (tid + 256 * j) * 4;
    float4 v = *(const float4*)&base[e];
    s[IDX(e)] = v.x; s[IDX(e + 1)] = v.y; s[IDX(e + 2)] = v.z; s[IDX(e + 3)] = v.w;
  }
  __syncthreads();
  for (int h = 1; h < 4096; h <<= 1) {
    for (int p = tid; p < 2048; p += 256) {
      int i0 = ((p & ~(h - 1)) << 1) | (p & (h - 1));
      int i1 = i0 + h;
      float a = s[IDX(i0)], b = s[IDX(i1)];
      s[IDX(i0)] = a + b; s[IDX(i1)] = a - b;
    }
    __syncthreads();
  }
  for (int j = 0; j < 4; ++j) {
    int e = (tid + 256 * j) * 4;
    float4 v;
    v.x = s[IDX(e)]; v.y = s[IDX(e + 1)]; v.z = s[IDX(e + 2)]; v.w = s[IDX(e + 3)];
    *(float4*)&base[e] = v;
  }
}

// ---------------------------------------------------------------------------
// Kernel 4: 6 FWHT stages over rows spaced `stride` apart. Tile = 64 rows x 64
// contiguous cols (256B-per-row coalesced loads), butterflies in LDS (stride-65
// rows => conflict-free for power-of-2 row distances).
// ---------------------------------------------------------------------------
__global__ __launch_bounds__(256)
void fwht_strided_kernel(float* __restrict__ data, int stride) {
  __shared__ float s[64 * 65];
  const int tid = threadIdx.x;
  const int cb = stride >> 6;                    // column blocks per "hi" group
  const long long hi = blockIdx.x / cb;
  const int lb = blockIdx.x % cb;
  float* base = data + hi * (long long)stride * 64 + (long long)lb * 64;

  for (int j = 0; j < 4; ++j) {
    int f = tid + 256 * j;                       // 1024 float4 transfers
    int row = f >> 4, c = (f & 15) * 4;
    float4 v = *(const float4*)&base[(long long)row * stride + c];
    s[row * 65 + c]     = v.x;
    s[row * 65 + c + 1] = v.y;
    s[row * 65 + c + 2] = v.z;
    s[row * 65 + c + 3] = v.w;
  }
  __syncthreads();
  for (int h = 1; h < 64; h <<= 1) {
    for (int p = tid; p < 2048; p += 256) {
      int col = p & 63;
      int pr  = p >> 6;                          // 32 row-pairs
      int r0  = ((pr & ~(h - 1)) << 1) | (pr & (h - 1));
      int i0 = r0 * 65 + col, i1 = (r0 + h) * 65 + col;
      float a = s[i0], b = s[i1];
      s[i0] = a + b; s[i1] = a - b;
    }
    __syncthreads();
  }
  for (int j = 0; j < 4; ++j) {
    int f = tid + 256 * j;
    int row = f >> 4, c = (f & 15) * 4;
    float4 v;
    v.x = s[row * 65 + c];
    v.y = s[row * 65 + c + 1];
    v.z = s[row * 65 + c + 2];
    v.w = s[row * 65 + c + 3];
    *(float4*)&base[(long long)row * stride + c] = v;
  }
}

// ---------------------------------------------------------------------------
// Kernel 5: out = x @ (W_init + dW)^T + b, fp32 WMMA 16x16x4.
// Block = 256 threads = 8 waves; block owns N strip [n0, n0+16); each wave owns
// 32 M-rows as two 16x16 accumulators. W elements touch HBM exactly once.
// ---------------------------------------------------------------------------
__global__ __launch_bounds__(256)
void gemm_kernel(const float* __restrict__ x, const float* __restrict__ Wi,
                 const float* __restrict__ t, const float* __restrict__ btot,
                 const float* __restrict__ divW, float* __restrict__ out) {
  __shared__ float xs[BATCH * XSTR];   // 256 x 32 (+pad)
  __shared__ float ws[16 * XSTR];      // 16 x 32 (+pad), fused W_init + dW/div
  const int tid = threadIdx.x;
  const int n0 = blockIdx.x * 16;
  const float invDiv = 1.0f / divW[0];
  const int lane = tid & 31;
  const int wv   = tid >> 5;
  const int col  = lane & 15;
  const int hiL  = lane >> 4;
  const int m0   = wv * 32;
  v8f acc0 = {};
  v8f acc1 = {};

  for (int k0 = 0; k0 < IN_F; k0 += KT) {
    // stage x tile: 256 rows x 32 K (coalesced 128B per 8 lanes)
    #pragma unroll
    for (int j = 0; j < 8; ++j) {
      int f = tid + 256 * j;
      int row = f >> 3, c = (f & 7) * 4;
      float4 v = *(const float4*)&x[(size_t)row * IN_F + k0 + c];
      *(float4*)&xs[row * XSTR + c] = v;
    }
    // stage fused W tile: Wtot[n][k] = W_init[n][k] + t[n*4096+k]/divisor_W
    if (tid < 128) {
      int row = tid >> 3, c = (tid & 7) * 4;
      size_t g = (size_t)(n0 + row) * IN_F + k0 + c;
      float4 w = *(const float4*)&Wi[g];
      float4 d = *(const float4*)&t[g];
      float4 r;
      r.x = w.x + d.x * invDiv;
      r.y = w.y + d.y * invDiv;
      r.z = w.z + d.z * invDiv;
      r.w = w.w + d.w * invDiv;
      *(float4*)&ws[row * XSTR + c] = r;
      if (k0 + KT < IN_F) __builtin_prefetch(&Wi[g + KT], 0, 1);  // global_prefetch_b8
    }
    __syncthreads();

    // A 16x4: lane<16 -> M=lane, K={k,k+1}; lane>=16 -> M=lane-16, K={k+2,k+3}
    // B 4x16: lane<16 -> N=lane, K={k,k+1}; lane>=16 -> N=lane-16, K={k+2,k+3}
    #pragma unroll
    for (int kk = 0; kk < KT; kk += 4) {
      v2f b  = *(const v2f*)&ws[col * XSTR + kk + 2 * hiL];
      v2f a0 = *(const v2f*)&xs[(m0 + col) * XSTR + kk + 2 * hiL];
      v2f a1 = *(const v2f*)&xs[(m0 + 16 + col) * XSTR + kk + 2 * hiL];
      acc0 = __builtin_amdgcn_wmma_f32_16x16x4_f32(false, a0, false, b,
                                                   (short)0, acc0, false, false);
      acc1 = __builtin_amdgcn_wmma_f32_16x16x4_f32(false, a1, false, b,
                                                   (short)0, acc1, false, false);
    }
    __syncthreads();
  }

  // C/D layout: VGPR r, lane<16 -> M=r, N=lane ; lane>=16 -> M=r+8, N=lane-16
  float bn = btot[n0 + col];
  #pragma unroll
  for (int r = 0; r < 8; ++r) {
    out[(size_t)(m0 + r + 8 * hiL) * OUT_F + n0 + col]      = acc0[r] + bn;
    out[(size_t)(m0 + 16 + r + 8 * hiL) * OUT_F + n0 + col] = acc1[r] + bn;
  }
}

// ---------------------------------------------------------------------------
extern "C" void kernel_launch(void* const* d_in, const int* in_sizes, int n_in,
                              void* d_out, int out_size, void* d_ws, size_t ws_size,
                              hipStream_t stream) {
  const float* x    = (const float*)d_in[0];
  const float* ld   = (const float*)d_in[1];
  const float* Wi   = (const float*)d_in[2];
  const float* bi   = (const float*)d_in[3];
  const float* BBW  = (const float*)d_in[4];
  const float* GGW  = (const float*)d_in[5];
  const float* divW = (const float*)d_in[6];
  const float* BBb  = (const float*)d_in[7];
  const float* GGb  = (const float*)d_in[8];
  const float* divb = (const float*)d_in[9];
  const int*   PiW  = (const int*)d_in[10];
  const int*   Pib  = (const int*)d_in[11];
  float* out = (float*)d_out;

  float* t    = (float*)d_ws;        // 16M floats (64 MB) — dW * divisor_W
  float* uW   = t + LLW;             // 1024 floats
  float* btot = uW + 1024;           // 4096 floats

  seed_kernel<<<2, 256, 0, stream>>>(ld, BBW, uW, bi, BBb, GGb, divb, Pib, btot);
  gather_kernel<<<LLW / 1024, 256, 0, stream>>>(PiW, GGW, uW, t);
  fwht_contig_kernel<<<LLW / 4096, 256, 0, stream>>>(t);          // bits 0..11
  fwht_strided_kernel<<<LLW / 4096, 256, 0, stream>>>(t, 4096);   // bits 12..17
  fwht_strided_kernel<<<LLW / 4096, 256, 0, stream>>>(t, 262144); // bits 18..23
  gemm_kernel<<<OUT_F / 16, 256, 0, stream>>>(x, Wi, t, btot, divW, out);
}